// MyNGCF_45715631898814
// MI455X (gfx1250) — compile-verified
//
#include <hip/hip_runtime.h>

#define NUSERS 60000
#define NITEMS 40000
#define NODES  100000
#define DIM    64
#define NNZ_E  3200000
#define NLAYERS 3
#define BATCH  1024
#define NEG_SLOPE 0.2f

typedef __attribute__((ext_vector_type(2))) float v2f;
typedef __attribute__((ext_vector_type(8))) float v8f;

// ---- build x0 = concat(embed_user, embed_item), float4-vectorized -------
__global__ void concat_kernel(const float4* __restrict__ eu,
                              const float4* __restrict__ ei,
                              float4* __restrict__ x0) {
    int tid = blockIdx.x * blockDim.x + threadIdx.x;
    const int totalU4 = NUSERS * DIM / 4;
    const int total4  = NODES * DIM / 4;
    if (tid >= total4) return;
    x0[tid] = (tid < totalU4) ? eu[tid] : ei[tid - totalU4];
}

__global__ void zero_kernel(float4* __restrict__ p, int n4) {
    int tid = blockIdx.x * blockDim.x + threadIdx.x;
    if (tid < n4) p[tid] = make_float4(0.f, 0.f, 0.f, 0.f);
}

// ---- SpMM: lie[row] += val * x[col]  (64 lanes per edge, coalesced) -----
// e is wave-uniform (each wave32 covers half the dims of one edge), so the
// index/value loads are scalarized via readfirstlane.
__global__ void spmm_kernel(const int*   __restrict__ row,
                            const int*   __restrict__ col,
                            const float* __restrict__ val,
                            const float* __restrict__ x,
                            float*       __restrict__ lie) {
    int t = blockIdx.x * blockDim.x + threadIdx.x;
    int e = __builtin_amdgcn_readfirstlane(t >> 6);   // wave-uniform edge id
    int d = t & 63;                                   // feature dim
    int   r = row[e];
    int   c = col[e];
    float v = val[e];
    unsafeAtomicAdd(&lie[(size_t)r * DIM + d], v * x[(size_t)c * DIM + d]);
}

// ---- fused dual-GEMM + bias + leaky_relu using V_WMMA_F32_16X16X4_F32 ---
// x_out = leaky_relu( lie @ W1 + b1 + (x .* lie) @ W2 + b2 )
// one wave -> one 16x64 output tile; 8 waves / block -> 128 rows / block.
// W1/W2 staged in LDS, pair-interleaved so each B fragment is one b64 load.
__global__ __launch_bounds__(256) void gemm_layer_kernel(
        const float* __restrict__ lie, const float* __restrict__ x,
        const float* __restrict__ W1,  const float* __restrict__ W2,
        const float* __restrict__ b1,  const float* __restrict__ b2,
        float* __restrict__ xout)
{
    // interleaved layout: element (k, n) at (k>>1)*128 + n*2 + (k&1)
    __shared__ float w1s[DIM / 2 * 2 * DIM];   // 4096 floats = 16 KB
    __shared__ float w2s[DIM / 2 * 2 * DIM];   // 4096 floats = 16 KB

    const int tid = threadIdx.x;
    #pragma unroll
    for (int i = 0; i < 16; ++i) {
        int idx = tid + i * 256;               // source linear index k*64+n
        int k   = idx >> 6;
        int n   = idx & 63;
        int dst = ((k >> 1) << 7) + (n << 1) + (k & 1);
        w1s[dst] = W1[idx];
        w2s[dst] = W2[idx];
    }
    __syncthreads();

    const int lane    = tid & 31;
    const int wave    = tid >> 5;
    const int m0      = (blockIdx.x * 8 + wave) * 16;
    if (m0 >= NODES) return;                   // NODES % 16 == 0: all-or-nothing tile

    const int halfsel = lane >> 4;             // 0: lanes 0-15, 1: lanes 16-31
    const int l       = lane & 15;
    const int kshift  = halfsel * 2;           // A holds K=0,1 (lo) / K=2,3 (hi)

    v8f acc[4];
    #pragma unroll
    for (int n = 0; n < 4; ++n)
        #pragma unroll
        for (int i = 0; i < 8; ++i) acc[n][i] = 0.0f;

    const int arow = m0 + l;                   // in-range by construction
    const float* lrow = lie + (size_t)arow * DIM;
    const float* xrow = x   + (size_t)arow * DIM;

    #pragma unroll 4
    for (int k0 = 0; k0 < DIM; k0 += 4) {
        const int kb = k0 + kshift;            // even
        v2f a_lie = *(const v2f*)(lrow + kb);  // 8B-aligned
        v2f xv    = *(const v2f*)(xrow + kb);
        v2f a_xl  = a_lie * xv;
        const int pbase = (kb >> 1) << 7;

        #pragma unroll
        for (int n = 0; n < 4; ++n) {
            const int ncol = n * 16 + l;
            v2f bw1 = *(const v2f*)(w1s + pbase + (ncol << 1));  // ds_load_b64
            v2f bw2 = *(const v2f*)(w2s + pbase + (ncol << 1));
            acc[n] = __builtin_amdgcn_wmma_f32_16x16x4_f32(
                         false, a_lie, false, bw1, (short)0, acc[n], false, false);
            acc[n] = __builtin_amdgcn_wmma_f32_16x16x4_f32(
                         false, a_xl,  false, bw2, (short)0, acc[n], false, false);
        }
    }

    // epilogue: bias + leaky relu, no predication (tile fully in range)
    const int rbase = m0 + halfsel * 8;
    #pragma unroll
    for (int n = 0; n < 4; ++n) {
        const int   ncol = n * 16 + l;
        const float bs   = b1[ncol] + b2[ncol];
        #pragma unroll
        for (int v = 0; v < 8; ++v) {
            float s = acc[n][v] + bs;
            xout[(size_t)(rbase + v) * DIM + ncol] = (s > 0.0f) ? s : NEG_SLOPE * s;
        }
    }
}

// ---- final gather of concat(x0..x3) rows --------------------------------
__global__ void gather_kernel(const float* __restrict__ x0,
                              const float* __restrict__ x1,
                              const float* __restrict__ x2,
                              const float* __restrict__ x3,
                              const int* __restrict__ su,
                              const int* __restrict__ obs,
                              const int* __restrict__ unobs,
                              float* __restrict__ out) {
    int tid = blockIdx.x * blockDim.x + threadIdx.x;
    const int total = 3 * BATCH * 4 * DIM;
    if (tid >= total) return;
    int sec = tid / (BATCH * 4 * DIM);
    int j   = (tid / (4 * DIM)) % BATCH;
    int c   = tid & (4 * DIM - 1);
    int node;
    if      (sec == 0) node = su[j];
    else if (sec == 1) node = NUSERS + obs[j];
    else               node = NUSERS + unobs[j];
    const float* src;
    switch (c >> 6) {
        case 0:  src = x0; break;
        case 1:  src = x1; break;
        case 2:  src = x2; break;
        default: src = x3; break;
    }
    out[tid] = src[(size_t)node * DIM + (c & 63)];
}

extern "C" void kernel_launch(void* const* d_in, const int* in_sizes, int n_in,
                              void* d_out, int out_size, void* d_ws, size_t ws_size,
                              hipStream_t stream) {
    const int*   edge_row   = (const int*)  d_in[0];
    const int*   edge_col   = (const int*)  d_in[1];
    const float* edge_val   = (const float*)d_in[2];
    const float* embed_user = (const float*)d_in[3];
    const float* embed_item = (const float*)d_in[4];
    const float* W1         = (const float*)d_in[5];
    const float* W2         = (const float*)d_in[6];
    const float* b1         = (const float*)d_in[7];
    const float* b2         = (const float*)d_in[8];
    const int*   su         = (const int*)  d_in[9];
    const int*   obs        = (const int*)  d_in[10];
    const int*   unobs      = (const int*)  d_in[11];
    float* out = (float*)d_out;

    const size_t xelem = (size_t)NODES * DIM;   // 6.4M floats
    float* ws = (float*)d_ws;
    float* xb[4];
    for (int i = 0; i < 4; ++i) xb[i] = ws + i * xelem;
    float* lie = ws + 4 * xelem;

    const int n4 = (int)(xelem / 4);            // 1.6M float4
    concat_kernel<<<(n4 + 255) / 256, 256, 0, stream>>>(
        (const float4*)embed_user, (const float4*)embed_item, (float4*)xb[0]);

    for (int i = 0; i < NLAYERS; ++i) {
        zero_kernel<<<(n4 + 255) / 256, 256, 0, stream>>>((float4*)lie, n4);
        // NNZ_E edges * 64 lanes = 204.8M threads -> 800000 blocks (exact)
        spmm_kernel<<<(NNZ_E * 64) / 256, 256, 0, stream>>>(
            edge_row, edge_col, edge_val, xb[i], lie);
        gemm_layer_kernel<<<(NODES + 127) / 128, 256, 0, stream>>>(
            lie, xb[i],
            W1 + (size_t)i * DIM * DIM, W2 + (size_t)i * DIM * DIM,
            b1 + (size_t)i * DIM,       b2 + (size_t)i * DIM,
            xb[i + 1]);
    }

    gather_kernel<<<(3 * BATCH * 4 * DIM) / 256, 256, 0, stream>>>(
        xb[0], xb[1], xb[2], xb[3], su, obs, unobs, out);
}